// DecoderLSTM_39453569581307
// MI455X (gfx1250) — compile-verified
//
#include <hip/hip_runtime.h>

// Problem dimensions (fixed by the reference).
#define B_   32
#define T_   128
#define H_   1024
#define V_   32000
#define G4H  (4 * H_)   // 4096 gate columns
#define K2H  (2 * H_)   // 2048 input features (concat)
#define BT   (B_ * T_)  // 4096 (batch*time rows)
#define LDSROW 1028     // W_hh LDS row pitch (floats), padded to kill bank conflicts

typedef float v2f __attribute__((ext_vector_type(2)));
typedef float v4f __attribute__((ext_vector_type(4)));
typedef float v8f __attribute__((ext_vector_type(8)));

// D = A(16x4 f32) * B(4x16 f32) + C(16x16 f32) on the CDNA5 matrix pipe.
__device__ __forceinline__ v8f wmma_f32_16x16x4(v2f a, v2f b, v8f c) {
  return __builtin_amdgcn_wmma_f32_16x16x4_f32(
      /*neg_a=*/false, a, /*neg_b=*/false, b,
      /*c_mod=*/(short)0, c, /*reuse_a=*/false, /*reuse_b=*/false);
}

// Store one 16x16 C tile (per-lane: 8 rows, 1 col) with a column bias added.
__device__ __forceinline__ void store_c_tile(float* __restrict__ dst, long long ld,
                                             long long row0, long long col,
                                             float bias, v8f acc) {
#pragma unroll
  for (int r = 0; r < 8; ++r) dst[(row0 + r) * ld + col] = acc[r] + bias;
}

__device__ __forceinline__ v4f relu4(v4f v) {
  v.x = fmaxf(v.x, 0.0f); v.y = fmaxf(v.y, 0.0f);
  v.z = fmaxf(v.z, 0.0f); v.w = fmaxf(v.w, 0.0f);
  return v;
}

// ---------------------------------------------------------------------------
// Phase 1: G1[bt, 4H] = relu(concat(enc[b], emb[tok(b,t)])) @ W_ih^T + (b_ih+b_hh)
// Grid: (BT/32, 4H/256), block 128 (4 waves). Wave = 32x64 output (2 M x 4 N
// tiles). b128 fragment loads cover two K-steps (.xy / .zw sub-fragments).
// ---------------------------------------------------------------------------
__global__ __launch_bounds__(128) void k_gates(
    const float* __restrict__ enc, const int* __restrict__ target,
    const float* __restrict__ emb, const float* __restrict__ W_ih,
    const float* __restrict__ b_ih, const float* __restrict__ b_hh,
    float* __restrict__ G1) {
  const int lane = threadIdx.x & 31;
  const int wave = threadIdx.x >> 5;
  const int half = lane >> 4;
  const int l16  = lane & 15;
  const int m0    = blockIdx.x * 32;
  const int nbase = (blockIdx.y * 16 + wave * 4) * 16;

  // A rows synthesized from enc/emb (token gather hoisted out of the K loop).
  const float* aEnc[2];
  const float* aEmb[2];
#pragma unroll
  for (int mt = 0; mt < 2; ++mt) {
    const int row = m0 + mt * 16 + l16;     // bt row
    const int b = row >> 7;                 // / T_
    const int t = row & (T_ - 1);
    const int tok = (t == 0) ? 1 : target[b * T_ + (t - 1)];
    aEnc[mt] = enc + (long long)b * H_;
    aEmb[mt] = emb + (long long)tok * H_;
  }
  const float* bRow[4];
#pragma unroll
  for (int nt = 0; nt < 4; ++nt)
    bRow[nt] = W_ih + (long long)(nbase + nt * 16 + l16) * K2H;

  v8f acc[2][4];
  const v8f zacc = {};
#pragma unroll
  for (int mt = 0; mt < 2; ++mt)
#pragma unroll
    for (int nt = 0; nt < 4; ++nt) acc[mt][nt] = zacc;

  for (int k = 0; k < K2H; k += 8) {
    const int kk = k + 4 * half;  // whole v4 stays on one side of the concat
    v4f a[2];
#pragma unroll
    for (int mt = 0; mt < 2; ++mt) {
      v4f v = (kk < H_) ? *(const v4f*)(aEnc[mt] + kk)
                        : *(const v4f*)(aEmb[mt] + (kk - H_));
      a[mt] = relu4(v);
    }
    v4f bb[4];
#pragma unroll
    for (int nt = 0; nt < 4; ++nt) bb[nt] = *(const v4f*)(bRow[nt] + kk);
#pragma unroll
    for (int mt = 0; mt < 2; ++mt)
#pragma unroll
      for (int nt = 0; nt < 4; ++nt) {
        acc[mt][nt] = wmma_f32_16x16x4(a[mt].xy, bb[nt].xy, acc[mt][nt]);
        acc[mt][nt] = wmma_f32_16x16x4(a[mt].zw, bb[nt].zw, acc[mt][nt]);
      }
  }

#pragma unroll
  for (int mt = 0; mt < 2; ++mt) {
    const long long row0 = m0 + mt * 16 + half * 8;
#pragma unroll
    for (int nt = 0; nt < 4; ++nt) {
      const int c = nbase + nt * 16 + l16;
      store_c_tile(G1, G4H, row0, c, b_ih[c] + b_hh[c], acc[mt][nt]);
    }
  }
}

// ---------------------------------------------------------------------------
// Phase 2: the FULL T=128 scan in one persistent kernel. 64 WGs (co-resident),
// each owning 16 h-columns. Each WG stages its 64x1024 W_hh slice into LDS
// ONCE (256 KB -- CDNA5's 320 KB/WGP makes this possible) and reuses it every
// step. Steps are separated by a software device-wide barrier (atomic counter
// + s_sleep spin + agent-scope acquire fence).
// ---------------------------------------------------------------------------
__global__ __launch_bounds__(128) void k_lstm_scan(
    const float* __restrict__ enc, const float* __restrict__ G1,
    const float* __restrict__ W_hh, float* __restrict__ Hout,
    float* __restrict__ c_ws, unsigned* __restrict__ bar) {
  __shared__ float sm_w[4][16][LDSROW];  // W_hh slice, padded rows
  __shared__ float gbuf[4][32][16];      // per-step gate pre-activations

  const int lane = threadIdx.x & 31;
  const int gate = threadIdx.x >> 5;  // wave id == gate index (i,f,g,o)
  const int half = lane >> 4;
  const int l16  = lane & 15;
  const int j0   = blockIdx.x * 16;

  // One-time cooperative stage of W_hh rows (gate*H + j0 + r) into LDS.
  for (int idx = threadIdx.x; idx < 64 * 256; idx += 128) {
    const int rrow = idx >> 8;         // 0..63
    const int c4   = (idx & 255) << 2; // float column
    const int g = rrow >> 4, r = rrow & 15;
    *(v4f*)(&sm_w[g][r][c4]) =
        *(const v4f*)(W_hh + (long long)(g * H_ + j0 + r) * H_ + c4);
  }
  __syncthreads();

  const float* __restrict__ bLds = &sm_w[gate][l16][0];

  for (int t = 0; t < T_; ++t) {
    const float* hb;
    long long astr;
    if (t == 0) { hb = enc;                          astr = H_; }
    else        { hb = Hout + (long long)(t - 1) * H_; astr = (long long)T_ * H_; }
    const float* aR0 = hb + (long long)(l16) * astr;       // batches 0..15
    const float* aR1 = hb + (long long)(16 + l16) * astr;  // batches 16..31

    v8f acc0 = {}, acc1 = {};
    for (int k = 0; k < H_; k += 8) {
      const int kk = k + 4 * half;
      v4f a0 = *(const v4f*)(aR0 + kk);
      v4f a1 = *(const v4f*)(aR1 + kk);
      v4f bb = *(const v4f*)(bLds + kk);  // ds_load_b128, conflict-free pitch
      acc0 = wmma_f32_16x16x4(a0.xy, bb.xy, acc0);
      acc1 = wmma_f32_16x16x4(a1.xy, bb.xy, acc1);
      acc0 = wmma_f32_16x16x4(a0.zw, bb.zw, acc0);
      acc1 = wmma_f32_16x16x4(a1.zw, bb.zw, acc1);
    }

#pragma unroll
    for (int r = 0; r < 8; ++r) {
      gbuf[gate][half * 8 + r][l16]      = acc0[r];
      gbuf[gate][16 + half * 8 + r][l16] = acc1[r];
    }
    __syncthreads();

    // Elementwise LSTM cell update: 32 batches x 16 columns.
    for (int e = threadIdx.x; e < 32 * 16; e += 128) {
      const int bb = e >> 4;
      const int jj = e & 15;
      const int j  = j0 + jj;
      const long long grow = (long long)(bb * T_ + t) * G4H;
      float iv = gbuf[0][bb][jj] + G1[grow + 0 * H_ + j];
      float fv = gbuf[1][bb][jj] + G1[grow + 1 * H_ + j];
      float gv = gbuf[2][bb][jj] + G1[grow + 2 * H_ + j];
      float ov = gbuf[3][bb][jj] + G1[grow + 3 * H_ + j];
      iv = 1.0f / (1.0f + __expf(-iv));
      fv = 1.0f / (1.0f + __expf(-fv));
      gv = tanhf(gv);
      ov = 1.0f / (1.0f + __expf(-ov));
      const float cprev = (t == 0) ? enc[bb * H_ + j] : c_ws[bb * H_ + j];
      const float cn = fv * cprev + iv * gv;
      const float hn = ov * tanhf(cn);
      c_ws[bb * H_ + j] = cn;
      Hout[(long long)(bb * T_ + t) * H_ + j] = hn;
    }

    // ---- device-wide barrier between steps ----
    __threadfence();   // release this thread's Hout/c writes
    __syncthreads();
    if (threadIdx.x == 0) {
      atomicAdd(bar, 1u);
      const unsigned target = (unsigned)(t + 1) * gridDim.x;
      while (__hip_atomic_load(bar, __ATOMIC_RELAXED, __HIP_MEMORY_SCOPE_AGENT) < target)
        __builtin_amdgcn_s_sleep(2);
    }
    __syncthreads();
    __builtin_amdgcn_fence(__ATOMIC_ACQUIRE, "agent");  // drop stale L0 lines
  }
}

// ---------------------------------------------------------------------------
// Phase 3: logits[bt, V] = Hout @ W_out^T + b_out (dominant GEMM, ~268 GFLOP).
// Grid: (BT/32, V/256), block 128; wave = 32x64 output. W_out (131 MB) fits in
// the 192 MB L2, so cross-tile re-reads stay on-chip.
// ---------------------------------------------------------------------------
__global__ __launch_bounds__(128) void k_logits(
    const float* __restrict__ Hout, const float* __restrict__ W_out,
    const float* __restrict__ b_out, float* __restrict__ out) {
  const int lane = threadIdx.x & 31;
  const int wave = threadIdx.x >> 5;
  const int half = lane >> 4;
  const int l16  = lane & 15;
  const int m0    = blockIdx.x * 32;
  const int nbase = (blockIdx.y * 16 + wave * 4) * 16;

  const float* aRow[2];
#pragma unroll
  for (int mt = 0; mt < 2; ++mt)
    aRow[mt] = Hout + (long long)(m0 + mt * 16 + l16) * H_;
  const float* bRow[4];
#pragma unroll
  for (int nt = 0; nt < 4; ++nt)
    bRow[nt] = W_out + (long long)(nbase + nt * 16 + l16) * H_;

  v8f acc[2][4];
  const v8f zacc = {};
#pragma unroll
  for (int mt = 0; mt < 2; ++mt)
#pragma unroll
    for (int nt = 0; nt < 4; ++nt) acc[mt][nt] = zacc;

  for (int k = 0; k < H_; k += 8) {
    const int kk = k + 4 * half;
    v4f a[2];
#pragma unroll
    for (int mt = 0; mt < 2; ++mt) a[mt] = *(const v4f*)(aRow[mt] + kk);
    v4f bb[4];
#pragma unroll
    for (int nt = 0; nt < 4; ++nt) bb[nt] = *(const v4f*)(bRow[nt] + kk);
#pragma unroll
    for (int mt = 0; mt < 2; ++mt)
#pragma unroll
      for (int nt = 0; nt < 4; ++nt) {
        acc[mt][nt] = wmma_f32_16x16x4(a[mt].xy, bb[nt].xy, acc[mt][nt]);
        acc[mt][nt] = wmma_f32_16x16x4(a[mt].zw, bb[nt].zw, acc[mt][nt]);
      }
  }

#pragma unroll
  for (int mt = 0; mt < 2; ++mt) {
    const long long row0 = m0 + mt * 16 + half * 8;
#pragma unroll
    for (int nt = 0; nt < 4; ++nt) {
      const int c = nbase + nt * 16 + l16;
      store_c_tile(out, V_, row0, c, b_out[c], acc[mt][nt]);
    }
  }
}

// ---------------------------------------------------------------------------
// Phase 4: in-place log_softmax over V per row, fully b128-vectorized
// (pure bandwidth: ~1.6 GB total over 23.3 TB/s HBM).
// ---------------------------------------------------------------------------
__global__ __launch_bounds__(256) void k_logsoftmax(float* __restrict__ out) {
  __shared__ float red[256];
  float* p = out + (long long)blockIdx.x * V_;
  v4f* p4 = (v4f*)p;
  const int NV4 = V_ / 4;  // 8000

  float m = -INFINITY;
  for (int i = threadIdx.x; i < NV4; i += 256) {
    v4f v = p4[i];
    m = fmaxf(m, fmaxf(fmaxf(v.x, v.y), fmaxf(v.z, v.w)));
  }
  red[threadIdx.x] = m;
  __syncthreads();
  for (int s = 128; s > 0; s >>= 1) {
    if (threadIdx.x < s) red[threadIdx.x] = fmaxf(red[threadIdx.x], red[threadIdx.x + s]);
    __syncthreads();
  }
  const float rowmax = red[0];
  __syncthreads();

  float sum = 0.0f;
  for (int i = threadIdx.x; i < NV4; i += 256) {
    v4f v = p4[i];
    sum += __expf(v.x - rowmax) + __expf(v.y - rowmax) +
           __expf(v.z - rowmax) + __expf(v.w - rowmax);
  }
  red[threadIdx.x] = sum;
  __syncthreads();
  for (int s = 128; s > 0; s >>= 1) {
    if (threadIdx.x < s) red[threadIdx.x] += red[threadIdx.x + s];
    __syncthreads();
  }
  const float lse = rowmax + __logf(red[0]);

  for (int i = threadIdx.x; i < NV4; i += 256) {
    v4f v = p4[i];
    v.x -= lse; v.y -= lse; v.z -= lse; v.w -= lse;
    p4[i] = v;
  }
}

// ---------------------------------------------------------------------------
extern "C" void kernel_launch(void* const* d_in, const int* in_sizes, int n_in,
                              void* d_out, int out_size, void* d_ws, size_t ws_size,
                              hipStream_t stream) {
  (void)in_sizes; (void)n_in; (void)out_size; (void)ws_size;
  const float* enc   = (const float*)d_in[0];
  const int*   tgt   = (const int*)d_in[1];
  const float* emb   = (const float*)d_in[2];
  const float* W_ih  = (const float*)d_in[3];
  const float* W_hh  = (const float*)d_in[4];
  const float* b_ih  = (const float*)d_in[5];
  const float* b_hh  = (const float*)d_in[6];
  const float* W_out = (const float*)d_in[7];
  const float* b_out = (const float*)d_in[8];
  float* out = (float*)d_out;

  // Workspace: Hout [BT, H] + c [B, H] + barrier counter (~17 MB).
  // G1 [BT, 4H] (64 MB) reuses the front of d_out (524 MB); it is fully
  // consumed by phase 2 before phase 3 overwrites d_out with logits.
  float*    Hout = (float*)d_ws;
  float*    c_ws = Hout + (size_t)BT * H_;
  unsigned* bar  = (unsigned*)(c_ws + (size_t)B_ * H_);
  float*    G1   = out;

  k_gates<<<dim3(BT / 32, G4H / 256), dim3(128), 0, stream>>>(
      enc, tgt, emb, W_ih, b_ih, b_hh, G1);

  hipMemsetAsync(bar, 0, sizeof(unsigned), stream);  // graph-capturable

  k_lstm_scan<<<dim3(H_ / 16), dim3(128), 0, stream>>>(
      enc, G1, W_hh, Hout, c_ws, bar);

  k_logits<<<dim3(BT / 32, V_ / 256), dim3(128), 0, stream>>>(
      Hout, W_out, b_out, out);

  k_logsoftmax<<<dim3(BT), dim3(256), 0, stream>>>(out);
}